// LabelPropagation_8735963480514
// MI455X (gfx1250) — compile-verified
//
#include <hip/hip_runtime.h>
#include <stdint.h>

#define NN 100000
#define CC 64
#define EE 3200000
#define LAYERS 10
#define ALPHA 0.9f

// ---------------- CSR build helpers ----------------

__global__ void zero_i32(int* p, int n) {
    int i = blockIdx.x * blockDim.x + threadIdx.x;
    if (i < n) p[i] = 0;
}

__global__ void count_deg(const int* __restrict__ dst, int* __restrict__ deg) {
    int e = blockIdx.x * blockDim.x + threadIdx.x;
    if (e < EE) atomicAdd(&deg[dst[e]], 1);
}

__global__ void make_norm(const int* __restrict__ deg, float* __restrict__ nrm) {
    int i = blockIdx.x * blockDim.x + threadIdx.x;
    if (i < NN) {
        float d = (float)max(deg[i], 1);
        nrm[i] = 1.0f / sqrtf(d);
    }
}

// chunk = 1024 elements per block (256 thr x 4)
__global__ void scan_partials(const int* __restrict__ deg, int* __restrict__ bsum) {
    __shared__ int sm[256];
    int t = threadIdx.x, b = blockIdx.x;
    int base = b * 1024 + t * 4;
    int s = 0;
#pragma unroll
    for (int k = 0; k < 4; ++k) { int i = base + k; if (i < NN) s += deg[i]; }
    sm[t] = s;
    __syncthreads();
    for (int off = 128; off > 0; off >>= 1) {
        if (t < off) sm[t] += sm[t + off];
        __syncthreads();
    }
    if (t == 0) bsum[b] = sm[0];
}

__global__ void scan_top(int* bsum, int nb) {
    __shared__ int sm[128];
    int t = threadIdx.x;
    int v = (t < nb) ? bsum[t] : 0;
    sm[t] = v;
    __syncthreads();
    for (int off = 1; off < 128; off <<= 1) {
        int x = (t >= off) ? sm[t - off] : 0;
        __syncthreads();
        sm[t] += x;
        __syncthreads();
    }
    if (t < nb) bsum[t] = sm[t] - v;   // exclusive
}

__global__ void scan_write(const int* __restrict__ deg, const int* __restrict__ bsum,
                           int* __restrict__ row) {
    __shared__ int sm[256];
    int t = threadIdx.x, b = blockIdx.x;
    int base = b * 1024 + t * 4;
    int v[4]; int s = 0;
#pragma unroll
    for (int k = 0; k < 4; ++k) { int i = base + k; v[k] = (i < NN) ? deg[i] : 0; s += v[k]; }
    sm[t] = s;
    __syncthreads();
    int mine = s;
    for (int off = 1; off < 256; off <<= 1) {
        int x = (t >= off) ? sm[t - off] : 0;
        __syncthreads();
        sm[t] += x;
        __syncthreads();
    }
    int run = bsum[b] + (sm[t] - mine);
#pragma unroll
    for (int k = 0; k < 4; ++k) { int i = base + k; if (i < NN) row[i] = run; run += v[k]; }
    if (b == 0 && t == 0) row[NN] = EE;
}

__global__ void copy_i32(const int* __restrict__ a, int* __restrict__ b, int n) {
    int i = blockIdx.x * blockDim.x + threadIdx.x;
    if (i < n) b[i] = a[i];
}

__global__ void bucket_edges(const int* __restrict__ src, const int* __restrict__ dst,
                             int* __restrict__ cursor, int* __restrict__ esrc) {
    int e = blockIdx.x * blockDim.x + threadIdx.x;
    if (e < EE) {
        int pos = atomicAdd(&cursor[dst[e]], 1);
        esrc[pos] = src[e];
    }
}

// ---------------- seed: y0 = mask ? labels : 0 ----------------

__global__ void seed_y(const float4* __restrict__ labels4,
                       const unsigned char* __restrict__ mask,
                       float4* __restrict__ y4) {
    int i = blockIdx.x * blockDim.x + threadIdx.x;
    if (i < NN * CC / 4) {
        int rowi = i >> 4;                 // 16 float4 per 64-wide row
        float4 l = labels4[i];
        if (!mask[rowi]) l = make_float4(0.f, 0.f, 0.f, 0.f);
        y4[i] = l;
    }
}

// ---------------- hot kernel: one wave32 per destination node ----------------
// lane holds channels (2*lane, 2*lane+1). Edge ids for the node's CSR bucket
// are staged into LDS with double-buffered async-to-LDS loads (ASYNCcnt).
// Inner loop batches 4 independent source-row gathers for memory-level
// parallelism (4 s_loads of norm + 4 global_load_b64 in flight per wait).

__launch_bounds__(256)
__global__ void prop_layer(const float* __restrict__ yin,
                           float* __restrict__ yout,
                           const float* __restrict__ nrm,
                           const int* __restrict__ row,
                           const int* __restrict__ esrc,
                           const float* __restrict__ labels,
                           const unsigned char* __restrict__ mask) {
    __shared__ int ebuf[8][2][32];       // 8 waves x double buffer x 32 edge ids
    const int lane = threadIdx.x & 31;
    const int wv   = threadIdx.x >> 5;
    const int node = blockIdx.x * 8 + wv;
    if (node >= NN) return;              // wave-uniform exit

    const int e0 = row[node];
    const int e1 = row[node + 1];

    const unsigned lds0 = (unsigned)(uintptr_t)&ebuf[wv][0][lane];
    const unsigned lds1 = (unsigned)(uintptr_t)&ebuf[wv][1][lane];

    // prime buffer 0 with the first chunk of edge ids (async DMA to LDS)
    {
        int me = e0 + lane;
        if (me < e1) {
            const int* g = esrc + me;
            asm volatile("global_load_async_to_lds_b32 %0, %1, off"
                         :: "v"(lds0), "v"(g) : "memory");
        }
    }

    // hoist combine-stage operands so their latency overlaps the gather loop
    const float nd = nrm[node];
    const size_t obase = (size_t)node * CC + lane * 2;
    const float2 lab = *(const float2*)(labels + obase);
    const bool mk = mask[node] != 0;

    float accx = 0.f, accy = 0.f;
    int buf = 0;
    for (int e = e0; e < e1; e += 32) {
        const int nexte = e + 32;
        if (nexte < e1) {
            // prefetch next chunk into the other buffer, then wait for current
            int me = nexte + lane;
            if (me < e1) {
                const int* g = esrc + me;
                unsigned l = buf ? lds0 : lds1;
                asm volatile("global_load_async_to_lds_b32 %0, %1, off"
                             :: "v"(l), "v"(g) : "memory");
            }
            asm volatile("s_wait_asynccnt 0x1" ::: "memory");
        } else {
            asm volatile("s_wait_asynccnt 0x0" ::: "memory");
        }

        int sid = ebuf[wv][buf][lane];   // one ds_load_b32 per lane
        const int cnt = min(32, e1 - e);
        int j = 0;
        for (; j + 4 <= cnt; j += 4) {   // 4-way batched gathers (MLP)
            int s0 = __builtin_amdgcn_readlane(sid, j + 0);
            int s1 = __builtin_amdgcn_readlane(sid, j + 1);
            int s2 = __builtin_amdgcn_readlane(sid, j + 2);
            int s3 = __builtin_amdgcn_readlane(sid, j + 3);
            float n0 = nrm[s0];
            float n1 = nrm[s1];
            float n2 = nrm[s2];
            float n3 = nrm[s3];
            float2 v0 = ((const float2*)(yin + (size_t)s0 * CC))[lane];
            float2 v1 = ((const float2*)(yin + (size_t)s1 * CC))[lane];
            float2 v2 = ((const float2*)(yin + (size_t)s2 * CC))[lane];
            float2 v3 = ((const float2*)(yin + (size_t)s3 * CC))[lane];
            accx = fmaf(n0, v0.x, accx); accy = fmaf(n0, v0.y, accy);
            accx = fmaf(n1, v1.x, accx); accy = fmaf(n1, v1.y, accy);
            accx = fmaf(n2, v2.x, accx); accy = fmaf(n2, v2.y, accy);
            accx = fmaf(n3, v3.x, accx); accy = fmaf(n3, v3.y, accy);
        }
        for (; j < cnt; ++j) {           // remainder
            int s = __builtin_amdgcn_readlane(sid, j);
            float ns = nrm[s];
            float2 v = ((const float2*)(yin + (size_t)s * CC))[lane];
            accx = fmaf(ns, v.x, accx);
            accy = fmaf(ns, v.y, accy);
        }
        buf ^= 1;
    }

    float lx = 0.f, ly = 0.f;
    if (mk) {
        lx = (1.0f - ALPHA) * lab.x;
        ly = (1.0f - ALPHA) * lab.y;
    }
    float ox = lx + ALPHA * accx * nd;
    float oy = ly + ALPHA * accy * nd;
    ox = fminf(fmaxf(ox, 0.f), 1.f);
    oy = fminf(fmaxf(oy, 0.f), 1.f);
    *((float2*)(yout + obase)) = make_float2(ox, oy);
}

// ---------------- launcher ----------------

extern "C" void kernel_launch(void* const* d_in, const int* in_sizes, int n_in,
                              void* d_out, int out_size, void* d_ws, size_t ws_size,
                              hipStream_t stream) {
    const float*         labels = (const float*)d_in[0];
    const unsigned char* mask   = (const unsigned char*)d_in[1]; // jnp.bool_: 1 byte/elem
    const int*           src    = (const int*)d_in[2];
    const int*           dst    = (const int*)d_in[3];
    float* y0 = (float*)d_out;

    uint8_t* w = (uint8_t*)d_ws;
    auto alloc = [&](size_t bytes) -> uint8_t* {
        uint8_t* p = w;
        w += (bytes + 255) & ~(size_t)255;
        return p;
    };
    float* nrm    = (float*)alloc((size_t)NN * sizeof(float));
    int*   row    = (int*)  alloc((size_t)(NN + 1) * sizeof(int));
    int*   cursor = (int*)  alloc((size_t)NN * sizeof(int));
    int*   bsum   = (int*)  alloc(128 * sizeof(int));
    int*   esrc   = (int*)  alloc((size_t)EE * sizeof(int));
    float* y1     = (float*)alloc((size_t)NN * CC * sizeof(float));
    (void)ws_size; (void)in_sizes; (void)n_in; (void)out_size;

    dim3 b256(256);
    const int NB = (NN + 1023) / 1024;   // 98 scan blocks

    // CSR-by-destination build (once per call, deterministic work)
    zero_i32     <<<(NN + 255) / 256, b256, 0, stream>>>(cursor, NN);
    count_deg    <<<(EE + 255) / 256, b256, 0, stream>>>(dst, cursor);
    make_norm    <<<(NN + 255) / 256, b256, 0, stream>>>(cursor, nrm);
    scan_partials<<<NB, b256, 0, stream>>>(cursor, bsum);
    scan_top     <<<1, 128, 0, stream>>>(bsum, NB);
    scan_write   <<<NB, b256, 0, stream>>>(cursor, bsum, row);
    copy_i32     <<<(NN + 255) / 256, b256, 0, stream>>>(row, cursor, NN);
    bucket_edges <<<(EE + 255) / 256, b256, 0, stream>>>(src, dst, cursor, esrc);

    // seed y0 = where(mask, labels, 0) into d_out
    seed_y<<<(NN * CC / 4 + 255) / 256, b256, 0, stream>>>(
        (const float4*)labels, mask, (float4*)y0);

    // 10 propagation layers, ping-pong so the last write lands in d_out
    for (int layer = 0; layer < LAYERS; ++layer) {
        const float* yin  = (layer & 1) ? y1 : y0;
        float*       yout = (layer & 1) ? y0 : y1;
        prop_layer<<<(NN + 7) / 8, b256, 0, stream>>>(
            yin, yout, nrm, row, esrc, labels, mask);
    }
}